// NormalizedMutualInformationLoss_10325101380042
// MI455X (gfx1250) — compile-verified
//
#include <hip/hip_runtime.h>
#include <hip/hip_bf16.h>
#include <math.h>

// ---------------------------------------------------------------------------
// NormalizedMutualInformationLoss for MI455X (gfx1250, wave32).
//
// Shapes: x,y = (64,1,512,512) f32.  Downsample ::2,::2 -> 65536 samples/batch.
// Stage 1 (bandwidth-bound, ~2.8us @ 23.3TB/s): streaming loads + per-wave
//   LDS-privatized 24x24 histograms (ds_add_u32), block-reduce, global atomics.
// Stage 2: per-batch NMI epilogue; marginal sums computed with
//   v_wmma_f32_16x16x4_f32 (joint * ones / joint^T * ones) on wave 0.
//   The joint is staged zero-padded to 32x32 in LDS so every WMMA operand
//   load is branch-free (no exec-mask save/restore, ds_load_b64/2addr merge).
// Stage 3: mean over 64 batches, clip, negate.
//
// Approximation: the reference's fixed-key 1e-4 Gaussian dither (JAX Threefry)
// is omitted; it shifts ~0.1% of samples across bin edges (bin width 0.0417
// vs 5e-5 perturbation) and is not reproducible in-kernel.
// ---------------------------------------------------------------------------

#define NB            24
#define NB2           (NB * NB)          // 576 bins
#define PAD           32                 // padded joint stride (zero-filled)
#define BATCHES       64
#define IMG_ELEMS     (512 * 512)        // 262144 per batch
#define SAMP_PER_B    (256 * 256)        // 65536 per batch
#define BLK_PER_B     16
#define THREADS       256
#define WAVES_PER_BLK (THREADS / 32)     // 8 waves (wave32)
#define SAMP_PER_BLK  (SAMP_PER_B / BLK_PER_B)  // 4096
#define ITERS         (SAMP_PER_BLK / THREADS)  // 16

typedef __attribute__((ext_vector_type(2))) float v2f;
typedef __attribute__((ext_vector_type(8))) float v8f;

// ---------------------------------------------------------------------------
__global__ void zero_ws_kernel(unsigned* p, int n) {
    int i = blockIdx.x * blockDim.x + threadIdx.x;
    if (i < n) p[i] = 0u;
}

// ---------------------------------------------------------------------------
// Stage 1: joint histogram.  grid = BATCHES * BLK_PER_B blocks of 256.
__global__ void __launch_bounds__(THREADS)
joint_hist_kernel(const float* __restrict__ x, const float* __restrict__ y,
                  unsigned* __restrict__ gjoint) {
    __shared__ unsigned lh[WAVES_PER_BLK * NB2];   // 18432 B: private hist/wave

    const int tid  = threadIdx.x;
    const int wave = tid >> 5;
    const int b    = blockIdx.x / BLK_PER_B;
    const int blk  = blockIdx.x % BLK_PER_B;

    for (int i = tid; i < WAVES_PER_BLK * NB2; i += THREADS) lh[i] = 0u;
    __syncthreads();

    const int base = blk * SAMP_PER_BLK;
    const size_t boff = (size_t)b * IMG_ELEMS;

#pragma unroll 4
    for (int it = 0; it < ITERS; ++it) {
        const int s = base + it * THREADS + tid;     // sample id in [0,65536)
        const int r = s >> 8;                        // downsampled row 0..255
        const int c = s & 255;                       // downsampled col 0..255
        const size_t off = boff + (size_t)r * 1024 + (size_t)(c << 1);

        // Prefetch next iteration's row segment (gfx1250 global_prefetch_b8).
        if (it + 1 < ITERS) {
            __builtin_prefetch(x + off + 1024, 0, 0);
            __builtin_prefetch(y + off + 1024, 0, 0);
        }

        const float xv = x[off];
        const float yv = y[off];

        // v = clip((v+1)*0.5, 0.001, 0.999); bin = clip(floor(v*24), 0, 23)
        float vx = fminf(fmaxf((xv + 1.0f) * 0.5f, 0.001f), 0.999f);
        float vy = fminf(fmaxf((yv + 1.0f) * 0.5f, 0.001f), 0.999f);
        int bx = min(max((int)(vx * 24.0f), 0), NB - 1);
        int by = min(max((int)(vy * 24.0f), 0), NB - 1);

        atomicAdd(&lh[wave * NB2 + bx * NB + by], 1u);   // ds_add_u32
    }
    __syncthreads();

    // Reduce the 8 per-wave histograms, push to the per-batch global joint.
    for (int i = tid; i < NB2; i += THREADS) {
        unsigned sum = 0u;
#pragma unroll
        for (int w = 0; w < WAVES_PER_BLK; ++w) sum += lh[w * NB2 + i];
        if (sum) atomicAdd(&gjoint[b * NB2 + i], sum);   // global_atomic_add_u32
    }
}

// ---------------------------------------------------------------------------
// Stage 2: per-batch NMI.  grid = 64 blocks of 256.  Marginals via WMMA.
__global__ void __launch_bounds__(THREADS)
nmi_per_batch_kernel(const unsigned* __restrict__ gjoint,
                     float* __restrict__ gnmi) {
    __shared__ float sj[PAD * PAD];  // normalized joint, zero-padded to 32x32
    __shared__ float sxh[PAD];       // x marginal (entries 24..31 unused)
    __shared__ float syh[PAD];       // y marginal
    __shared__ float s_mi;

    const int b   = blockIdx.x;
    const int tid = threadIdx.x;

    // Zero the padded joint, then fill the live 24x24 region.
    for (int i = tid; i < PAD * PAD; i += THREADS) sj[i] = 0.0f;
    if (tid == 0) s_mi = 0.0f;
    __syncthreads();

    // Every sample lands in exactly one bin -> total is exactly 65536.
    const float inv_total = 1.0f / (65536.0f + 1e-10f);
    for (int i = tid; i < NB2; i += THREADS) {
        const int bi = i / NB, bj = i % NB;
        sj[bi * PAD + bj] = (float)gjoint[b * NB2 + i] * inv_total;
    }
    __syncthreads();

    // --- Marginals on wave 0 via v_wmma_f32_16x16x4_f32 -------------------
    // D(16x16) += A(16x4) x B(4x16), B = ones => D[m,n] = sum_k A[m,k].
    // A layout (32-bit 16x4): lanes 0-15 hold M=0..15 with K=0,1 in v0,v1;
    // lanes 16-31 hold K=2,3.  => lane (g=l>>4, m=l&15): A[j] = J[m][4kc+2g+j].
    // D layout (32-bit 16x16): vgpr r, lanes 0-15 -> M=r; lanes 16-31 -> M=r+8.
    // Padding rows/cols 24..31 are zero, so all tile reads are branch-free.
    if (tid < 32) {                      // whole wave 0 => EXEC all ones
        const int g = tid >> 4, m = tid & 15;
        v2f ones2; ones2[0] = 1.0f; ones2[1] = 1.0f;   // layout-invariant B
        v8f ax0 = {}, ax1 = {}, ay0 = {}, ay1 = {};

#pragma unroll
        for (int kc = 0; kc < 6; ++kc) { // K = 0..23 in chunks of 4
            const int k0 = kc * 4 + g * 2;
            v2f a;
            // x_hist rows 0..15:  A = joint[m][k]      (ds_load_b64)
            a[0] = sj[m * PAD + k0]; a[1] = sj[m * PAD + k0 + 1];
            ax0 = __builtin_amdgcn_wmma_f32_16x16x4_f32(
                      false, a, false, ones2, (short)0, ax0, false, false);
            // x_hist rows 16..31: A = joint[m+16][k]   (rows >=24 are zero)
            a[0] = sj[(m + 16) * PAD + k0]; a[1] = sj[(m + 16) * PAD + k0 + 1];
            ax1 = __builtin_amdgcn_wmma_f32_16x16x4_f32(
                      false, a, false, ones2, (short)0, ax1, false, false);
            // y_hist rows 0..15:  A = joint^T[m][k] = joint[k][m]
            a[0] = sj[k0 * PAD + m]; a[1] = sj[(k0 + 1) * PAD + m];
            ay0 = __builtin_amdgcn_wmma_f32_16x16x4_f32(
                      false, a, false, ones2, (short)0, ay0, false, false);
            // y_hist rows 16..31  (cols >=24 are zero)
            a[0] = sj[k0 * PAD + m + 16]; a[1] = sj[(k0 + 1) * PAD + m + 16];
            ay1 = __builtin_amdgcn_wmma_f32_16x16x4_f32(
                      false, a, false, ones2, (short)0, ay1, false, false);
        }

        // Column N=0 lives in lanes 0 (rows 0-7) and 16 (rows 8-15).
        if (m == 0) {
#pragma unroll
            for (int r = 0; r < 8; ++r) {
                const int row = r + 8 * g;
                sxh[row]      = ax0[r];
                syh[row]      = ay0[r];
                sxh[row + 16] = ax1[r];   // rows 24..31 are harmless zeros
                syh[row + 16] = ay1[r];
            }
        }
    }
    __syncthreads();

    // --- MI sum over 576 cells --------------------------------------------
    float part = 0.0f;
    for (int i = tid; i < NB2; i += THREADS) {
        const int bi = i / NB, bj = i % NB;
        const float je = sj[bi * PAD + bj] + 1e-5f;
        const float lp = logf((sxh[bi] + 1e-5f) * (syh[bj] + 1e-5f));
        part += je * (logf(je) - lp);
    }
    atomicAdd(&s_mi, part);              // ds_add_f32
    __syncthreads();

    if (tid == 0) {
        float hx = 0.0f, hy = 0.0f;
#pragma unroll
        for (int i = 0; i < NB; ++i) {
            const float xe = sxh[i] + 1e-5f;
            const float ye = syh[i] + 1e-5f;
            hx -= xe * logf(xe);
            hy -= ye * logf(ye);
        }
        const float se = hx + hy;
        float nmi = (se < 1e-10f) ? 0.0f : 2.0f * s_mi / se;
        nmi = fminf(fmaxf(nmi, -1.0f), 1.0f);
        gnmi[b] = nmi;
    }
}

// ---------------------------------------------------------------------------
// Stage 3: loss = -clip(mean(nmi), -1, 1).  1 block of 64 threads.
__global__ void reduce_mean_kernel(const float* __restrict__ gnmi,
                                   float* __restrict__ out) {
    __shared__ float s;
    if (threadIdx.x == 0) s = 0.0f;
    __syncthreads();
    if (threadIdx.x < BATCHES) atomicAdd(&s, gnmi[threadIdx.x]);
    __syncthreads();
    if (threadIdx.x == 0) {
        float m = s * (1.0f / (float)BATCHES);
        m = fminf(fmaxf(m, -1.0f), 1.0f);
        out[0] = -m;
    }
}

// ---------------------------------------------------------------------------
extern "C" void kernel_launch(void* const* d_in, const int* in_sizes, int n_in,
                              void* d_out, int out_size, void* d_ws, size_t ws_size,
                              hipStream_t stream) {
    (void)in_sizes; (void)n_in; (void)out_size; (void)ws_size;

    const float* x = (const float*)d_in[0];
    const float* y = (const float*)d_in[1];
    float* out = (float*)d_out;

    // Workspace layout: [64*576] u32 joint counts, then [64] f32 nmi values.
    unsigned* gjoint = (unsigned*)d_ws;
    float*    gnmi   = (float*)((char*)d_ws + (size_t)BATCHES * NB2 * sizeof(unsigned));

    const int zero_n = BATCHES * NB2 + BATCHES;   // nmi floats zero = bits 0
    zero_ws_kernel<<<(zero_n + THREADS - 1) / THREADS, THREADS, 0, stream>>>(
        gjoint, zero_n);

    joint_hist_kernel<<<BATCHES * BLK_PER_B, THREADS, 0, stream>>>(x, y, gjoint);

    nmi_per_batch_kernel<<<BATCHES, THREADS, 0, stream>>>(gjoint, gnmi);

    reduce_mean_kernel<<<1, 64, 0, stream>>>(gnmi, out);
}